// GCN_86457691668980
// MI455X (gfx1250) — compile-verified
//
#include <hip/hip_runtime.h>

#define N_NODES 100000
#define N_EDGES 1600000
#define D_IN    128
#define D_H     64   // d_h == d_out == 64

typedef __attribute__((ext_vector_type(2))) float v2f;
typedef __attribute__((ext_vector_type(8))) float v8f;

// ---------------------------------------------------------------------------
// Dense GEMM  C[nrows x 64] = act(A[nrows x K]) @ W[K x 64]  via f32 WMMA.
// One wave computes a 16-row strip x all 64 columns (4 16x16 tiles).
// f32 WMMA operand layout (ISA 7.12.2, 32-bit A 16x4):
//   lane 0-15 : M=lane,    holds K = k..k+1
//   lane16-31 : M=lane-16, holds K = k+2..k+3
// B (4x16) mirrors this with N striped across lanes.
// C/D: VGPR r, lanes 0-15 -> M=r, lanes 16-31 -> M=r+8, N=lane&15.
//
// W is staged in LDS as packed row-pairs  P[kp*64 + c] = {W[2kp][c], W[2kp+1][c]}
// so each B operand in the inner loop is one ds_load_b64.
// ---------------------------------------------------------------------------
template <int K, bool RELU_IN>
__global__ void gemm_wmma_kernel(const float* __restrict__ A,
                                 const float* __restrict__ W,
                                 float* __restrict__ C, int nrows) {
  __shared__ v2f Pw[(K / 2) * 64];

  // Cooperative LDS fill (coalesced global reads, packed-pair LDS layout).
  for (int idx = threadIdx.x; idx < K * 64; idx += blockDim.x) {
    const int k = idx >> 6;       // W row
    const int c = idx & 63;       // W col
    ((float*)Pw)[(k >> 1) * 128 + c * 2 + (k & 1)] = W[idx];
  }
  __syncthreads();

  const int lane = threadIdx.x & 31;
  const int wave = blockIdx.x * (blockDim.x >> 5) + (threadIdx.x >> 5);
  const int row0 = wave * 16;
  if (row0 >= nrows) return;            // wave-uniform: EXEC all-ones below

  const int half = lane >> 4;           // 0 or 1
  const int l15  = lane & 15;
  const long long arow = row0 + l15;

  v8f acc0 = {}, acc1 = {}, acc2 = {}, acc3 = {};

  for (int k = 0; k < K; k += 4) {
    const int kk = k + 2 * half;
    // A operand: 2 consecutive f32 of row `arow` at column kk (8B aligned)
    v2f a = *(const v2f*)(A + arow * K + kk);
    if (RELU_IN) {
      a.x = fmaxf(a.x, 0.0f);
      a.y = fmaxf(a.y, 0.0f);
    }
    // B operands from LDS: pair row kp = kk/2 = k/2 + half
    const v2f* __restrict__ Pk = Pw + (k / 2 + half) * 64;
    const v2f b0 = Pk[l15 +  0];
    const v2f b1 = Pk[l15 + 16];
    const v2f b2 = Pk[l15 + 32];
    const v2f b3 = Pk[l15 + 48];

    acc0 = __builtin_amdgcn_wmma_f32_16x16x4_f32(false, a, false, b0, (short)0, acc0, false, false);
    acc1 = __builtin_amdgcn_wmma_f32_16x16x4_f32(false, a, false, b1, (short)0, acc1, false, false);
    acc2 = __builtin_amdgcn_wmma_f32_16x16x4_f32(false, a, false, b2, (short)0, acc2, false, false);
    acc3 = __builtin_amdgcn_wmma_f32_16x16x4_f32(false, a, false, b3, (short)0, acc3, false, false);
  }

#pragma unroll
  for (int r = 0; r < 8; ++r) {
    const long long m = row0 + r + 8 * half;
    float* __restrict__ crow = C + m * 64;
    crow[l15 +  0] = acc0[r];
    crow[l15 + 16] = acc1[r];
    crow[l15 + 32] = acc2[r];
    crow[l15 + 48] = acc3[r];
  }
}

// ---------------------------------------------------------------------------
// out[n][j] = bias[j]  for all n  (fuses the "+ b" into the scatter init)
// ---------------------------------------------------------------------------
__global__ void init_bias_kernel(float* __restrict__ out,
                                 const float* __restrict__ bias, int total) {
  int tid = blockIdx.x * blockDim.x + threadIdx.x;
  if (tid < total) out[tid] = bias[tid & 63];
}

// ---------------------------------------------------------------------------
// SpMM scatter:  out[dst[e]] += w[e] * sup[src[e]]   (64 feats, f32)
// 16 threads per edge, float4 per thread -> half-wave reads one contiguous
// 256B row of sup and issues 4 global_atomic_add_f32 into one row of out.
// ---------------------------------------------------------------------------
__global__ void spmm_atomic_kernel(const float* __restrict__ sup,
                                   const int* __restrict__ src,
                                   const int* __restrict__ dst,
                                   const float* __restrict__ ew,
                                   float* __restrict__ out, int nedges) {
  long long tid = (long long)blockIdx.x * blockDim.x + threadIdx.x;
  int e = (int)(tid >> 4);
  if (e >= nedges) return;
  int j = ((int)tid & 15) * 4;

  float w = ew[e];
  long long s = src[e];
  long long d = dst[e];

  const float4 v = *(const float4*)(sup + s * 64 + j);
  float* __restrict__ o = out + d * 64 + j;
  atomicAdd(o + 0, w * v.x);
  atomicAdd(o + 1, w * v.y);
  atomicAdd(o + 2, w * v.z);
  atomicAdd(o + 3, w * v.w);
}

// ---------------------------------------------------------------------------
extern "C" void kernel_launch(void* const* d_in, const int* in_sizes, int n_in,
                              void* d_out, int out_size, void* d_ws, size_t ws_size,
                              hipStream_t stream) {
  const float* features = (const float*)d_in[0];  // [N,128]
  const int*   src      = (const int*)  d_in[1];  // [E]
  const int*   dst      = (const int*)  d_in[2];  // [E]
  const float* ew       = (const float*)d_in[3];  // [E]
  const float* W1       = (const float*)d_in[4];  // [128,64]
  const float* b1       = (const float*)d_in[5];  // [64]
  const float* W2       = (const float*)d_in[6];  // [64,64]
  const float* b2       = (const float*)d_in[7];  // [64]
  const float* W3       = (const float*)d_in[8];  // [64,64]
  const float* b3       = (const float*)d_in[9];  // [64]
  float* out = (float*)d_out;                     // [N,64]

  const long long feat_elems = (long long)N_NODES * D_H;  // 6.4M floats
  float* bufA = (float*)d_ws;                // support buffer
  float* bufB = bufA + feat_elems;           // aggregation buffer

  const dim3 blk(256);
  const int gemm_blocks = (N_NODES / 16 + 7) / 8;              // 8 waves/block
  const int init_blocks = (int)((feat_elems + 255) / 256);
  const int spmm_blocks = (int)(((long long)N_EDGES * 16 + 255) / 256);

  // ---- Layer 1: h1 = relu(spmm(X @ W1) + b1) -------------------------------
  gemm_wmma_kernel<D_IN, false><<<gemm_blocks, blk, 0, stream>>>(features, W1, bufA, N_NODES);
  init_bias_kernel<<<init_blocks, blk, 0, stream>>>(bufB, b1, (int)feat_elems);
  spmm_atomic_kernel<<<spmm_blocks, blk, 0, stream>>>(bufA, src, dst, ew, bufB, N_EDGES);

  // ---- Layer 2: h2 = relu(spmm(relu(h1) @ W2) + b2) ------------------------
  gemm_wmma_kernel<D_H, true><<<gemm_blocks, blk, 0, stream>>>(bufB, W2, bufA, N_NODES);
  init_bias_kernel<<<init_blocks, blk, 0, stream>>>(bufB, b2, (int)feat_elems);
  spmm_atomic_kernel<<<spmm_blocks, blk, 0, stream>>>(bufA, src, dst, ew, bufB, N_EDGES);

  // ---- Layer 3: out = spmm(relu(h2) @ W3) + b3 (no relu) -------------------
  gemm_wmma_kernel<D_H, true><<<gemm_blocks, blk, 0, stream>>>(bufB, W3, bufA, N_NODES);
  init_bias_kernel<<<init_blocks, blk, 0, stream>>>(out, b3, (int)feat_elems);
  spmm_atomic_kernel<<<spmm_blocks, blk, 0, stream>>>(bufA, src, dst, ew, out, N_EDGES);

  (void)in_sizes; (void)n_in; (void)out_size; (void)ws_size;
}